// VanillaSoftmax_20718922236516
// MI455X (gfx1250) — compile-verified
//
#include <hip/hip_runtime.h>

typedef __attribute__((ext_vector_type(2))) float        v2f;
typedef __attribute__((ext_vector_type(8))) float        v8f;
typedef __attribute__((ext_vector_type(4))) unsigned int v4u;
typedef __attribute__((ext_vector_type(8))) int          v8i;
typedef __attribute__((ext_vector_type(4))) int          v4i;

#define N_ROWS 4096
#define VOCAB  50000
#define DIM    64
#define NTILES (VOCAB / 16)     // 3125 column tiles
#define NWAVES 8
#define KM     4                // M-tiles per wave -> 64 rows per workgroup
#define BR     (KM * 16)
#define RB     (N_ROWS / BR)    // 64 row-blocks
#define VS     8                // V-splits -> grid RB x VS = 512 workgroups
#define CH     2                // E tiles per TDM chunk (2 x 4KB = 8KB)
#define EPSF   1e-7f

// workspace layout (float offsets)
#define WS1_OFF 0                       // [RB][VS][BR][2]  per-slice (m, s)
#define WS2_OFF (RB * VS * BR * 2)      // [N_ROWS]         per-row  M + ln(S)
#define WS3_OFF (WS2_OFF + N_ROWS)     // [RB][VS][BR]     per-slice clipped-prob sums

// ---- Tensor Data Mover: 1-D contiguous chunk of E -> LDS (async, TENSORcnt) ----
__device__ __forceinline__ void tdm_load_chunk(const float* gsrc, unsigned ldsOff,
                                               int nelem /* f32 elements */) {
  unsigned long long ga = (unsigned long long)(size_t)gsrc;
  v4u g0;
  g0[0] = 1u;                                        // count=1, user descriptor
  g0[1] = ldsOff;                                    // LDS byte address
  g0[2] = (unsigned)(ga & 0xFFFFFFFFu);              // global_addr[31:0]
  g0[3] = (unsigned)((ga >> 32) & 0x1FFFFFFu)        // global_addr[56:32]
        | (2u << 30);                                // type = 2 ("image")
  v8i g1;
  g1[0] = 0x20000;                                   // data_size=2 (4B), wg_mask=0
  g1[1] = (int)(((unsigned)nelem & 0xFFFFu) << 16);  // tensor_dim0[15:0]
  g1[2] = (int)(((unsigned)nelem >> 16) | (1u << 16)); // dim0[31:16], tensor_dim1=1
  g1[3] = (int)((unsigned)nelem << 16);              // tile_dim0 = nelem
  g1[4] = 1;                                         // tile_dim1 = 1
  g1[5] = (int)nelem;                                // tensor_dim0_stride
  g1[6] = 0;
  g1[7] = 0;
  v4i z4 = {0, 0, 0, 0};                             // groups 2/3: tensors <= 2D
  v8i z8 = {0, 0, 0, 0, 0, 0, 0, 0};
  __builtin_amdgcn_tensor_load_to_lds(g0, g1, z4, z4, z8, 0);
}

// -------- pass 1: per (row-block, V-slice) online row max / sum --------
__global__ __launch_bounds__(256) void pass1_maxsum(
    const float* __restrict__ inputs, const float* __restrict__ emb,
    float* __restrict__ ws)
{
  const int tid  = threadIdx.x;
  const int wave = tid >> 5, lane = tid & 31;
  const int nsel = lane & 15, hi = lane >> 4, koff = hi << 1;
  const int rb = blockIdx.x, vs = blockIdx.y;
  const int rowBase = rb * BR;
  const int t0 = (vs * NTILES) / VS, t1 = ((vs + 1) * NTILES) / VS;
  // contiguous tile slice per wave (so TDM chunks are contiguous)
  const int w0 = t0 + ((t1 - t0) * wave) / NWAVES;
  const int w1 = t0 + ((t1 - t0) * (wave + 1)) / NWAVES;

  __shared__ __align__(16) float ebuf[NWAVES][2][CH * 16 * DIM]; // 128 KB
  __shared__ float mbuf[NWAVES][BR];
  __shared__ float sbuf[NWAVES][BR];

  // A fragments for KM row-tiles (16x4 f32 WMMA layout):
  // lane<16: row=lane, K={4kk,4kk+1}; lane>=16: row=lane-16, K={4kk+2,4kk+3}
  v2f a[KM][16];
  #pragma unroll
  for (int mt = 0; mt < KM; ++mt) {
    const float* arow = inputs + (size_t)(rowBase + mt * 16 + nsel) * DIM;
    #pragma unroll
    for (int kk = 0; kk < 16; ++kk)
      a[mt][kk] = *(const v2f*)(arow + 4 * kk + koff);
  }

  float m[KM][8], s[KM][8];
  #pragma unroll
  for (int mt = 0; mt < KM; ++mt)
    #pragma unroll
    for (int j = 0; j < 8; ++j) { m[mt][j] = -__builtin_inff(); s[mt][j] = 0.f; }

  const int nch = (w1 - w0 + CH - 1) / CH;
  if (nch > 0) {
    int cn0 = min(CH, w1 - w0);
    tdm_load_chunk(emb + (size_t)w0 * 16 * DIM,
                   (unsigned)(size_t)&ebuf[wave][0][0], cn0 * 16 * DIM);
  }
  for (int ch = 0; ch < nch; ++ch) {
    const int cs = w0 + ch * CH;
    const int cn = min(CH, w1 - cs);
    if (ch + 1 < nch) {                 // prefetch next chunk into other buffer
      const int ns = w0 + (ch + 1) * CH;
      const int nn = min(CH, w1 - ns);
      asm volatile("s_wait_dscnt 0x0" ::: "memory");   // WAR vs old LDS reads
      tdm_load_chunk(emb + (size_t)ns * 16 * DIM,
                     (unsigned)(size_t)&ebuf[wave][(ch + 1) & 1][0], nn * 16 * DIM);
      __builtin_amdgcn_s_wait_tensorcnt((short)1);     // current chunk resident
    } else {
      __builtin_amdgcn_s_wait_tensorcnt((short)0);
    }
    const float* lb = &ebuf[wave][ch & 1][0];
    for (int i = 0; i < cn; ++i) {
      const float* brow = lb + (i * 16 + nsel) * DIM;
      v8f c[KM];
      #pragma unroll
      for (int mt = 0; mt < KM; ++mt) c[mt] = {};
      #pragma unroll
      for (int kk = 0; kk < 16; ++kk) {        // B frag from LDS feeds KM wmmas
        v2f b = *(const v2f*)(brow + 4 * kk + koff);
        #pragma unroll
        for (int mt = 0; mt < KM; ++mt)
          c[mt] = __builtin_amdgcn_wmma_f32_16x16x4_f32(false, a[mt][kk], false, b,
                                                        (short)0, c[mt], false, false);
      }
      #pragma unroll
      for (int mt = 0; mt < KM; ++mt)
        #pragma unroll
        for (int j = 0; j < 8; ++j) {          // c[j]: row = j + 8*hi, col = nsel
          float x = c[mt][j];
          float d = x - m[mt][j];
          float e = __expf(-__builtin_fabsf(d));
          if (d > 0.f) { s[mt][j] = __builtin_fmaf(s[mt][j], e, 1.f); m[mt][j] = x; }
          else         { s[mt][j] += e; }
        }
    }
  }

  // reduce (m,s) across the 16 lanes sharing each row-half
  #pragma unroll
  for (int off = 1; off < 16; off <<= 1)
    #pragma unroll
    for (int mt = 0; mt < KM; ++mt)
      #pragma unroll
      for (int j = 0; j < 8; ++j) {
        float mo = __shfl_xor(m[mt][j], off);
        float so = __shfl_xor(s[mt][j], off);
        float mn = fmaxf(m[mt][j], mo);
        s[mt][j] = s[mt][j] * __expf(m[mt][j] - mn) + so * __expf(mo - mn);
        m[mt][j] = mn;
      }
  if (nsel == 0)
    #pragma unroll
    for (int mt = 0; mt < KM; ++mt)
      #pragma unroll
      for (int j = 0; j < 8; ++j) {
        mbuf[wave][mt * 16 + (hi << 3) + j] = m[mt][j];
        sbuf[wave][mt * 16 + (hi << 3) + j] = s[mt][j];
      }
  __syncthreads();
  if (tid < BR) {                  // combine the 8 waves, emit per-slice (m,s)
    float M = -__builtin_inff(), S = 0.f;
    #pragma unroll
    for (int w = 0; w < NWAVES; ++w) {
      float mo = mbuf[w][tid], so = sbuf[w][tid];
      float mn = fmaxf(M, mo);
      S = S * __expf(M - mn) + so * __expf(mo - mn);
      M = mn;
    }
    float* o = ws + WS1_OFF + ((size_t)(rb * VS + vs) * BR + tid) * 2;
    o[0] = M; o[1] = S;
  }
}

// -------- pass 1b: combine V-slices -> per-row (M + ln S) --------
__global__ __launch_bounds__(256) void combine_maxsum(float* __restrict__ ws)
{
  const int row = blockIdx.x * 256 + threadIdx.x;   // 4096 rows
  const int rb = row / BR, r = row % BR;
  float M = -__builtin_inff(), S = 0.f;
  #pragma unroll
  for (int vs = 0; vs < VS; ++vs) {
    const float* p = ws + WS1_OFF + ((size_t)(rb * VS + vs) * BR + r) * 2;
    float mo = p[0], so = p[1];
    float mn = fmaxf(M, mo);
    S = S * __expf(M - mn) + so * __expf(mo - mn);
    M = mn;
  }
  ws[WS2_OFF + row] = M + __logf(S);
}

// -------- pass 2: recompute logits, write probs, clipped-prob sums --------
// Softmax shift folded into the GEMM as a 17th K-step: extra wmma with
// A = -(M+lnS) on K-lane 0, B = e0  =>  c = x - (M+lnS), so p = exp(c).
__global__ __launch_bounds__(256) void pass2_probs(
    const float* __restrict__ inputs, const float* __restrict__ emb,
    float* __restrict__ ws, float* __restrict__ probs)
{
  const int tid  = threadIdx.x;
  const int wave = tid >> 5, lane = tid & 31;
  const int nsel = lane & 15, hi = lane >> 4, koff = hi << 1;
  const int rb = blockIdx.x, vs = blockIdx.y;
  const int rowBase = rb * BR;
  const int t0 = (vs * NTILES) / VS, t1 = ((vs + 1) * NTILES) / VS;
  const int w0 = t0 + ((t1 - t0) * wave) / NWAVES;
  const int w1 = t0 + ((t1 - t0) * (wave + 1)) / NWAVES;

  __shared__ __align__(16) float ebuf[NWAVES][2][CH * 16 * DIM]; // 128 KB
  __shared__ float scbuf[NWAVES][BR];

  v2f a[KM][16];
  #pragma unroll
  for (int mt = 0; mt < KM; ++mt) {
    const float* arow = inputs + (size_t)(rowBase + mt * 16 + nsel) * DIM;
    #pragma unroll
    for (int kk = 0; kk < 16; ++kk)
      a[mt][kk] = *(const v2f*)(arow + 4 * kk + koff);
  }
  v2f bext = {};
  v2f aM[KM];
  #pragma unroll
  for (int mt = 0; mt < KM; ++mt) aM[mt] = (v2f){};
  if (hi == 0) {
    bext[0] = 1.0f;                       // B[K=64][n] = 1
    #pragma unroll
    for (int mt = 0; mt < KM; ++mt)       // A[row][K=64] = -(M + lnS)
      aM[mt][0] = -ws[WS2_OFF + rowBase + mt * 16 + nsel];
  }

  float sc[KM][8];
  #pragma unroll
  for (int mt = 0; mt < KM; ++mt)
    #pragma unroll
    for (int j = 0; j < 8; ++j) sc[mt][j] = 0.f;

  const int nch = (w1 - w0 + CH - 1) / CH;
  if (nch > 0) {
    int cn0 = min(CH, w1 - w0);
    tdm_load_chunk(emb + (size_t)w0 * 16 * DIM,
                   (unsigned)(size_t)&ebuf[wave][0][0], cn0 * 16 * DIM);
  }
  for (int ch = 0; ch < nch; ++ch) {
    const int cs = w0 + ch * CH;
    const int cn = min(CH, w1 - cs);
    if (ch + 1 < nch) {
      const int ns = w0 + (ch + 1) * CH;
      const int nn = min(CH, w1 - ns);
      asm volatile("s_wait_dscnt 0x0" ::: "memory");
      tdm_load_chunk(emb + (size_t)ns * 16 * DIM,
                     (unsigned)(size_t)&ebuf[wave][(ch + 1) & 1][0], nn * 16 * DIM);
      __builtin_amdgcn_s_wait_tensorcnt((short)1);
    } else {
      __builtin_amdgcn_s_wait_tensorcnt((short)0);
    }
    const float* lb = &ebuf[wave][ch & 1][0];
    for (int i = 0; i < cn; ++i) {
      const float* brow = lb + (i * 16 + nsel) * DIM;
      v8f c[KM];
      #pragma unroll
      for (int mt = 0; mt < KM; ++mt) c[mt] = {};
      #pragma unroll
      for (int kk = 0; kk < 16; ++kk) {
        v2f b = *(const v2f*)(brow + 4 * kk + koff);
        #pragma unroll
        for (int mt = 0; mt < KM; ++mt)
          c[mt] = __builtin_amdgcn_wmma_f32_16x16x4_f32(false, a[mt][kk], false, b,
                                                        (short)0, c[mt], false, false);
      }
      #pragma unroll
      for (int mt = 0; mt < KM; ++mt)       // fold-in -(M+lnS)
        c[mt] = __builtin_amdgcn_wmma_f32_16x16x4_f32(false, aM[mt], false, bext,
                                                      (short)0, c[mt], false, false);
      const int col = ((cs + i) << 4) + nsel;
      #pragma unroll
      for (int mt = 0; mt < KM; ++mt)
        #pragma unroll
        for (int j = 0; j < 8; ++j) {
          float p = __expf(c[mt][j]);       // already shifted & normalized
          probs[(size_t)(rowBase + mt * 16 + (hi << 3) + j) * VOCAB + col] = p;
          sc[mt][j] += fminf(fmaxf(p, EPSF), 1.0f - EPSF);
        }
    }
  }

  #pragma unroll
  for (int off = 1; off < 16; off <<= 1)
    #pragma unroll
    for (int mt = 0; mt < KM; ++mt)
      #pragma unroll
      for (int j = 0; j < 8; ++j) sc[mt][j] += __shfl_xor(sc[mt][j], off);
  if (nsel == 0)
    #pragma unroll
    for (int mt = 0; mt < KM; ++mt)
      #pragma unroll
      for (int j = 0; j < 8; ++j)
        scbuf[wave][mt * 16 + (hi << 3) + j] = sc[mt][j];
  __syncthreads();
  if (tid < BR) {
    float SC = 0.f;
    #pragma unroll
    for (int w = 0; w < NWAVES; ++w) SC += scbuf[w][tid];
    ws[WS3_OFF + (size_t)(rb * VS + vs) * BR + tid] = SC;
  }
}

// -------- finalize: loss = mean(log(sum_clip) - log(clip(p[label]))) --------
__global__ __launch_bounds__(1024) void finalize_loss(
    const float* __restrict__ ws, const float* __restrict__ probs,
    const int* __restrict__ label, float* __restrict__ out)
{
  const int tid = threadIdx.x;
  float acc = 0.f;
  for (int row = tid; row < N_ROWS; row += 1024) {
    const int rb = row / BR, r = row % BR;
    float SC = 0.f;
    #pragma unroll
    for (int vs = 0; vs < VS; ++vs)
      SC += ws[WS3_OFF + (size_t)(rb * VS + vs) * BR + r];
    float p  = probs[(size_t)row * VOCAB + label[row]];
    float pc = fminf(fmaxf(p, EPSF), 1.0f - EPSF);
    acc += __logf(SC) - __logf(pc);
  }
  #pragma unroll
  for (int off = 1; off < 32; off <<= 1) acc += __shfl_xor(acc, off);
  __shared__ float buf[32];
  if ((tid & 31) == 0) buf[tid >> 5] = acc;
  __syncthreads();
  if (tid == 0) {
    float t = 0.f;
    for (int w = 0; w < 32; ++w) t += buf[w];
    out[(size_t)N_ROWS * VOCAB] = t / (float)N_ROWS;
  }
}

extern "C" void kernel_launch(void* const* d_in, const int* in_sizes, int n_in,
                              void* d_out, int out_size, void* d_ws, size_t ws_size,
                              hipStream_t stream) {
  (void)in_sizes; (void)n_in; (void)out_size; (void)ws_size;
  const int*   label  = (const int*)d_in[0];
  const float* inputs = (const float*)d_in[1];
  const float* emb    = (const float*)d_in[2];
  float* out = (float*)d_out;
  float* ws  = (float*)d_ws;     // ~400 KB used

  dim3 grid(RB, VS);
  pass1_maxsum  <<<grid, 32 * NWAVES, 0, stream>>>(inputs, emb, ws);
  combine_maxsum<<<N_ROWS / 256, 256, 0, stream>>>(ws);
  pass2_probs   <<<grid, 32 * NWAVES, 0, stream>>>(inputs, emb, ws, out);
  finalize_loss <<<1, 1024, 0, stream>>>(ws, out /* probs */, label, out);
}